// Embedding_44719199486126
// MI455X (gfx1250) — compile-verified
//
#include <hip/hip_runtime.h>

// Embedding gather: out[r, :] = table[ids[r], :]  (r = 0..BATCH*HIST-1, DIM=32)
// Pure data movement: ~180 MB HBM traffic, ~8 us floor at 23.3 TB/s.
// CDNA5 paths used: async global->LDS id staging (ASYNCcnt), global_prefetch_b8,
// coalesced global_load_b128 gathers, non-temporal b128 stores.

#define THREADS        256
#define ROWS_PER_BLOCK 256
#define DIM            32
#define CHUNKS_PER_ROW 8   // 32 floats / 4-float chunks

typedef __attribute__((ext_vector_type(4))) float v4f;
typedef __attribute__((address_space(1))) int  as1_int;
typedef __attribute__((address_space(3))) int  as3_int;

__global__ __launch_bounds__(THREADS)
void embed_gather_kernel(const int* __restrict__ ids,
                         const float* __restrict__ table,
                         float* __restrict__ out,
                         int nrows) {
    __shared__ int s_ids[ROWS_PER_BLOCK];

    const int t    = threadIdx.x;
    const int base = blockIdx.x * ROWS_PER_BLOCK;
    const int r    = base + t;

    // ---- Stage this block's 256 ids into LDS ----------------------------
#if __has_builtin(__builtin_amdgcn_global_load_async_to_lds_b32)
    if (r < nrows) {
        __builtin_amdgcn_global_load_async_to_lds_b32(
            (as1_int*)(const_cast<int*>(ids) + r),
            (as3_int*)(&s_ids[t]),
            /*offset=*/0, /*cpol=*/0);
    }
  #if __has_builtin(__builtin_amdgcn_s_wait_asynccnt)
    __builtin_amdgcn_s_wait_asynccnt(0);
  #else
    asm volatile("s_wait_asynccnt 0x0" ::: "memory");
  #endif
#else
    s_ids[t] = (r < nrows) ? ids[r] : 0;
#endif
    __syncthreads();   // all waves have drained their async writes to LDS

    // ---- Prefetch the gathered table rows (global_prefetch_b8) ----------
    if (r < nrows) {
        const float* rowp = table + (size_t)s_ids[t] * DIM;
        __builtin_prefetch(rowp,      0, 3);   // first 64B of the 128B row
        __builtin_prefetch(rowp + 16, 0, 3);   // second 64B
    }

    // ---- Gather: 8 lanes per row, one float4 (b128) per lane ------------
    // Lanes i..i+7 read contiguous 128B of one table row -> one coalesced
    // request per row. Output stores are non-temporal so the 105MB stream
    // does not evict the L2-resident 128MB table.
#pragma unroll
    for (int k = 0; k < CHUNKS_PER_ROW; ++k) {
        const int i     = k * THREADS + t;        // 0..2047 within the tile
        const int row   = i >> 3;                 // 0..255
        const int chunk = i & 7;                  // 0..7
        const int grow  = base + row;
        if (grow < nrows) {
            const int id = s_ids[row];
            const v4f* src = (const v4f*)(table + (size_t)id * DIM) + chunk;
            v4f v = *src;                         // global_load_b128
            v4f* dst = (v4f*)(out + (size_t)grow * DIM) + chunk;
            __builtin_nontemporal_store(v, dst);  // global_store_b128 (NT)
        }
    }
}

extern "C" void kernel_launch(void* const* d_in, const int* in_sizes, int n_in,
                              void* d_out, int out_size, void* d_ws, size_t ws_size,
                              hipStream_t stream) {
    const int*   ids   = (const int*)d_in[0];    // [BATCH*HIST] flat
    const float* table = (const float*)d_in[1];  // [VOCAB, DIM]
    float*       out   = (float*)d_out;          // [BATCH*HIST, DIM]

    const int nrows  = in_sizes[0];              // 4096*200 = 819200
    const int blocks = (nrows + ROWS_PER_BLOCK - 1) / ROWS_PER_BLOCK;

    embed_gather_kernel<<<blocks, THREADS, 0, stream>>>(ids, table, out, nrows);
}